// GraphConvolutionAttentivePooling_11304353923438
// MI455X (gfx1250) — compile-verified
//
#include <hip/hip_runtime.h>
#include <hip/hip_bf16.h>
#include <math.h>

// ---------------------------------------------------------------------------
// GraphConvolutionAttentivePooling for MI455X (gfx1250)
//   h  = leaky_relu( e @ (x @ W0^T + b0) )
//   h  = tanh( e @ (h @ W1^T + b1) )
//   a  = h @ w2^T + b2
//   w  = softmax(mnm * a + mask, axis=1)        [M, N]
//   out = (w @ h  [M,64],  w  [M,N])
//
// Big cost: streaming e (1 GiB fp32) twice -> bandwidth floor ~92 us total.
// fp32 WMMA (V_WMMA_F32_16X16X4_F32) is used for the e-GEMMs: fp32 inputs,
// memory-bound op, K=4 fp32 matrix pipe is adequate vs the HBM roofline.
// ---------------------------------------------------------------------------

#define NN 16384   // nodes
#define MM 512     // molecules
#define DD 64      // feature dim (in == out)
#define SPLITK 4   // deterministic split-K for the big GEMM

typedef float v2f __attribute__((ext_vector_type(2)));
typedef float v8f __attribute__((ext_vector_type(8)));

__device__ __forceinline__ v8f wmma_f32(v2f a, v2f b, v8f c) {
  // 8-arg pattern: (neg_a, A, neg_b, B, c_mod, C, reuse_a, reuse_b)
  return __builtin_amdgcn_wmma_f32_16x16x4_f32(false, a, false, b,
                                               (short)0, c, false, false);
}

// ---------------------------------------------------------------------------
// y[n,o] = sum_d x[n,d] * W[o,d] + b[o]     (tiny, VALU is fine)
// ---------------------------------------------------------------------------
__global__ __launch_bounds__(256) void lin64(const float* __restrict__ x,
                                             const float* __restrict__ W,
                                             const float* __restrict__ b,
                                             float* __restrict__ y) {
  int idx = blockIdx.x * 256 + threadIdx.x;   // n*64 + o, grid covers N*64
  int n = idx >> 6, o = idx & 63;
  const float* xr = x + (size_t)n * DD;
  const float* wr = W + (size_t)o * DD;
  float acc = b[o];
#pragma unroll
  for (int d = 0; d < DD; ++d) acc = fmaf(xr[d], wr[d], acc);
  y[idx] = acc;
}

// ---------------------------------------------------------------------------
// Partial GEMM: P[s][n][o] = e[n, s*4096:(s+1)*4096] @ B[s*4096:(s+1)*4096, o]
// One wave = 16 rows x 64 cols (4 WMMA f32 16x16x4 accumulator tiles).
// B panel (64 K x 64 cols) staged in LDS transposed (pitch 66 -> conflict-free
// stores, 8B-aligned fragment loads).  e read with non-temporal hint.
// ---------------------------------------------------------------------------
__global__ __launch_bounds__(128) void gemm_e(const float* __restrict__ e,
                                              const float* __restrict__ B,
                                              float* __restrict__ P) {
  __shared__ __align__(16) float ldsB[64 * 66];
  const int tid  = threadIdx.x;
  const int wave = tid >> 5;
  const int lane = tid & 31;
  const int m    = lane & 15;      // M (A) / N (B,C) index within tile
  const int half = lane >> 4;      // selects K pair
  const int r0   = blockIdx.x * 64 + wave * 16;
  const int k0   = blockIdx.y * (NN / SPLITK);

  v8f acc0 = {}, acc1 = {}, acc2 = {}, acc3 = {};
  const float* erow = e + (size_t)(r0 + m) * NN + 2 * half;

  for (int kb = 0; kb < NN / SPLITK; kb += 64) {
    __syncthreads();   // protect previous panel's LDS reads
    // stage B[k0+kb .. +63][0..63] transposed: ldsB[c*66 + kk]
    for (int i = tid; i < 64 * 64; i += 128) {
      int kk = i >> 6, c = i & 63;
      ldsB[c * 66 + kk] = B[(size_t)(k0 + kb + kk) * DD + c];
    }
    __syncthreads();

    const float* ep = erow + k0 + kb;
#pragma unroll 4
    for (int kk = 0; kk < 64; kk += 4) {
      // A 16x4 fp32 fragment: lanes 0-15 K={kk,kk+1}, lanes 16-31 K={kk+2,kk+3}
      v2f a = __builtin_nontemporal_load((const v2f*)(ep + kk));
      const float* bp = &ldsB[m * 66 + kk + 2 * half];   // 8B aligned
      v2f b0 = *(const v2f*)(bp);
      v2f b1 = *(const v2f*)(bp + 16 * 66);
      v2f b2 = *(const v2f*)(bp + 32 * 66);
      v2f b3 = *(const v2f*)(bp + 48 * 66);
      acc0 = wmma_f32(a, b0, acc0);
      acc1 = wmma_f32(a, b1, acc1);
      acc2 = wmma_f32(a, b2, acc2);
      acc3 = wmma_f32(a, b3, acc3);
    }
  }

  // C/D layout: VGPR v, lane -> row r0 + v + 8*half, col = tile*16 + m
  float* pr = P + (size_t)blockIdx.y * NN * DD;
#pragma unroll
  for (int v = 0; v < 8; ++v) {
    int row = r0 + v + 8 * half;
    float* o = pr + (size_t)row * DD + m;
    o[0]  = acc0[v];
    o[16] = acc1[v];
    o[32] = acc2[v];
    o[48] = acc3[v];
  }
}

// ---------------------------------------------------------------------------
// Deterministic split-K reduction + activation.  act: 0 = leaky_relu(0.01),
// 1 = tanh.
// ---------------------------------------------------------------------------
__global__ __launch_bounds__(256) void reduce_act(const float* __restrict__ P,
                                                  float* __restrict__ y,
                                                  int act) {
  size_t idx = (size_t)blockIdx.x * 256 + threadIdx.x;   // grid covers N*64
  float v = P[idx];
  v += P[idx + (size_t)1 * NN * DD];
  v += P[idx + (size_t)2 * NN * DD];
  v += P[idx + (size_t)3 * NN * DD];
  if (act == 0) v = v >= 0.0f ? v : 0.01f * v;
  else          v = tanhf(v);
  y[idx] = v;
}

// ---------------------------------------------------------------------------
// a[n] = h[n,:] . w2 + b2
// ---------------------------------------------------------------------------
__global__ __launch_bounds__(256) void attn64(const float* __restrict__ h,
                                              const float* __restrict__ w2,
                                              const float* __restrict__ b2,
                                              float* __restrict__ a) {
  int n = blockIdx.x * 256 + threadIdx.x;   // grid covers N
  const float* hr = h + (size_t)n * DD;
  float acc = b2[0];
#pragma unroll
  for (int d = 0; d < DD; ++d) acc = fmaf(hr[d], w2[d], acc);
  a[n] = acc;
}

// ---------------------------------------------------------------------------
// Per-molecule masked softmax over nodes + weighted pooling.
// logits = mnm*a + mask ; w = softmax(logits) ; pooled = w @ h
// One 256-thread block per molecule; fixed-order reductions (deterministic).
// ---------------------------------------------------------------------------
__global__ __launch_bounds__(256) void pool_softmax(
    const float* __restrict__ a, const float* __restrict__ mnm,
    const float* __restrict__ mask, const float* __restrict__ h,
    float* __restrict__ pooled, float* __restrict__ wgt) {
  const int mrow = blockIdx.x, tid = threadIdx.x;
  __shared__ float red[256];
  const float* mn = mnm  + (size_t)mrow * NN;
  const float* mk = mask + (size_t)mrow * NN;

  // pass 1: row max
  float lm = -INFINITY;
  for (int n = tid; n < NN; n += 256) {
    float l = mn[n] * a[n] + mk[n];
    lm = fmaxf(lm, l);
  }
  red[tid] = lm; __syncthreads();
  for (int s = 128; s > 0; s >>= 1) {
    if (tid < s) red[tid] = fmaxf(red[tid], red[tid + s]);
    __syncthreads();
  }
  float rowmax = red[0]; __syncthreads();

  // pass 2: sum of exp
  float ls = 0.0f;
  for (int n = tid; n < NN; n += 256) {
    float l = mn[n] * a[n] + mk[n];
    ls += expf(l - rowmax);
  }
  red[tid] = ls; __syncthreads();
  for (int s = 128; s > 0; s >>= 1) {
    if (tid < s) red[tid] += red[tid + s];
    __syncthreads();
  }
  float inv = 1.0f / red[0]; __syncthreads();

  // pass 3: write normalized weights
  float* wr = wgt + (size_t)mrow * NN;
  for (int n = tid; n < NN; n += 256) {
    float l = mn[n] * a[n] + mk[n];
    wr[n] = expf(l - rowmax) * inv;
  }
  __syncthreads();

  // pass 4: pooled[m, f] = sum_n w[n] * h[n, f]
  const int f = tid & 63, g = tid >> 6;   // 4 partial groups per feature
  float acc = 0.0f;
  for (int n = g; n < NN; n += 4) acc = fmaf(wr[n], h[(size_t)n * DD + f], acc);
  red[tid] = acc; __syncthreads();
  if (tid < 64)
    pooled[(size_t)mrow * DD + tid] =
        red[tid] + red[tid + 64] + red[tid + 128] + red[tid + 192];
}

// ---------------------------------------------------------------------------
extern "C" void kernel_launch(void* const* d_in, const int* in_sizes, int n_in,
                              void* d_out, int out_size, void* d_ws,
                              size_t ws_size, hipStream_t stream) {
  const float* nf   = (const float*)d_in[0];   // [N,64]
  const float* e    = (const float*)d_in[1];   // [N,N]
  const float* mnm  = (const float*)d_in[2];   // [M,N]
  const float* mask = (const float*)d_in[3];   // [M,N]
  const float* W0   = (const float*)d_in[4];
  const float* b0   = (const float*)d_in[5];
  const float* W1   = (const float*)d_in[6];
  const float* b1   = (const float*)d_in[7];
  const float* w2   = (const float*)d_in[8];
  const float* b2   = (const float*)d_in[9];

  float* out    = (float*)d_out;
  float* pooled = out;                       // [M,64]
  float* wgt    = out + (size_t)MM * DD;     // [M,N]
  // Reuse the 32 MB weight region of d_out as split-K partial scratch
  // (needs SPLITK*N*64 = 4M floats <= M*N = 8.4M floats); it is fully
  // overwritten by pool_softmax afterwards.
  float* pbuf = wgt;

  float* buf0 = (float*)d_ws;                      // [N,64]
  float* buf1 = buf0 + (size_t)NN * DD;            // [N,64]
  float* avec = buf1 + (size_t)NN * DD;            // [N]

  dim3 gGemm(NN / 64, SPLITK);

  // layer 1
  lin64<<<NN * DD / 256, 256, 0, stream>>>(nf, W0, b0, buf0);
  gemm_e<<<gGemm, 128, 0, stream>>>(e, buf0, pbuf);
  reduce_act<<<NN * DD / 256, 256, 0, stream>>>(pbuf, buf1, 0);   // leaky_relu
  // layer 2
  lin64<<<NN * DD / 256, 256, 0, stream>>>(buf1, W1, b1, buf0);
  gemm_e<<<gGemm, 128, 0, stream>>>(e, buf0, pbuf);
  reduce_act<<<NN * DD / 256, 256, 0, stream>>>(pbuf, buf1, 1);   // tanh
  // attention + pooling
  attn64<<<NN / 256, 256, 0, stream>>>(buf1, w2, b2, avec);
  pool_softmax<<<MM, 256, 0, stream>>>(avec, mnm, mask, buf1, pooled, wgt);
}